// TwoLayerTransformer_49606872269422
// MI455X (gfx1250) — compile-verified
//
#include <hip/hip_runtime.h>

#define EPSN 1e-7f
#define T_   2048
#define TP   2049   // T+1
#define D_   64
#define H_   8
#define B_   8
#define PADL 128    // lower zero-guard for Toeplitz table (dist < 0 region)
#define PADH 64     // upper zero-guard (rows past t=2048 in the tail tile)
#define EWSZ (PADL + TP + PADH)

typedef __attribute__((ext_vector_type(2))) float v2f;
typedef __attribute__((ext_vector_type(8))) float v8f;

// ---------------------------------------------------------------------------
// Kernel 1: per-head exp(W) table + inclusive-prefix-sum reciprocal (invZ).
// eW[h][d] = exp(W[d,h]) for d<2048, eW[h][2048] = 1.0 (the dist==T corner).
// invZ[h][t] = 1 / sum_{d=0..t} eW[h][d]   (softmax denominators, causal).
// ---------------------------------------------------------------------------
__global__ __launch_bounds__(256)
void toeplitz_precompute(const float* __restrict__ W,
                         float* __restrict__ eW,
                         float* __restrict__ invZ) {
    __shared__ float segsum[256];
    __shared__ float segoff[257];
    const int h    = blockIdx.x;
    const int tid  = threadIdx.x;
    const int base = tid * 8;

    float e[8];
    float s = 0.0f;
#pragma unroll
    for (int j = 0; j < 8; ++j) {
        float w = W[(base + j) * H_ + h];
        e[j] = expf(w);
        eW[h * TP + base + j] = e[j];
        s += e[j];
    }
    segsum[tid] = s;
    __syncthreads();
    if (tid == 0) {
        float run = 0.0f;
        for (int i = 0; i < 256; ++i) { segoff[i] = run; run += segsum[i]; }
        segoff[256] = run;
    }
    __syncthreads();

    float run = segoff[tid];
#pragma unroll
    for (int j = 0; j < 8; ++j) {
        run += e[j];                       // inclusive prefix
        invZ[h * TP + base + j] = 1.0f / run;
    }
    if (tid == 0) {
        eW[h * TP + T_]   = 1.0f;                       // exp(0) corner
        invZ[h * TP + T_] = 1.0f / (segoff[256] + 1.0f);
    }
}

// ---------------------------------------------------------------------------
// Kernel 2: causal Toeplitz attention + double l2norm, fp32 WMMA (16x16x4).
// Grid: (65 t-tiles, B, H). Block: 256 threads = 8 waves.
// Wave w: tsub = w>>2 (two 16-row t-subtiles), dsub = w&3 (four 16-col d-subtiles).
// The eW table is zero-padded in LDS so causal masking is just a table read:
// the k-loop is a fixed, fully unrolled 16 steps with unconditional LDS loads,
// staged into registers ahead of the 16-deep WMMA accumulation chain.
// ---------------------------------------------------------------------------
__global__ __launch_bounds__(256)
void toeplitz_attn(const float* __restrict__ x,
                   const float* __restrict__ eW,
                   const float* __restrict__ invZ,
                   float* __restrict__ out) {
    __shared__ __align__(16) float lds_ew[EWSZ];      // padded Toeplitz table
    __shared__ __align__(16) float lds_x[64 * D_];    // 16 KB: 64-row xp chunk
    __shared__ __align__(16) float lds_out[32 * 68];  // padded stride 68 for banks
    __shared__ float lds_scale[32];

    const int t0   = blockIdx.x * 32;
    const int b    = blockIdx.y;
    const int h    = blockIdx.z;
    const int tid  = threadIdx.x;
    const int lane = tid & 31;
    const int wave = tid >> 5;
    const int tsub = wave >> 2;          // 0..1
    const int dsub = wave & 3;           // 0..3
    const int half = lane >> 4;          // lane group 0 / 1
    const int l15  = lane & 15;

    // Stage padded eW table: pad regions read exact 0.0 (causal mask / tail).
    for (int i = tid; i < EWSZ; i += 256) {
        const int d = i - PADL;
        lds_ew[i] = (d >= 0 && d < TP) ? eW[h * TP + d] : 0.0f;
    }

    v8f acc = {};

    const int trow  = t0 + tsub * 16 + l15;              // this lane's A row
    const int k0    = half * 2;                          // K pair per lane half
    const int bcol  = dsub * 16 + l15;                   // this lane's B column
    const int s_end = (t0 + 31 < T_) ? (t0 + 31) : T_;   // causal upper bound

    for (int sc = 0; sc <= s_end; sc += 64) {
        __syncthreads();   // protect lds_x reuse + (first iter) lds_ew fill
        // Stage 64 rows of xp[b] (rows >= 2048 are zeros) as float4.
        {
            const float4* x4 = (const float4*)x;
            for (int i = tid; i < 64 * (D_ / 4); i += 256) {
                int r  = i >> 4;           // row within chunk
                int c4 = i & 15;           // float4 column
                int s  = sc + r;
                float4 v = make_float4(0.f, 0.f, 0.f, 0.f);
                if (s < T_) v = x4[((size_t)b * T_ + s) * (D_ / 4) + c4];
                ((float4*)lds_x)[i] = v;
            }
        }
        __syncthreads();

        // Stage all A pairs (adjacent table entries) and B pairs to registers.
        v2f A[16], Bv[16];
        const int idx = PADL + (trow - sc - k0) - 1;     // a.y index at kc=0
#pragma unroll
        for (int kc = 0; kc < 16; ++kc) {
            A[kc].y = lds_ew[idx - 4 * kc];              // dist = trow-s-k0-1
            A[kc].x = lds_ew[idx - 4 * kc + 1];          // dist = trow-s-k0
            const int rl = kc * 4 + k0;
            Bv[kc].x = lds_x[ rl      * D_ + bcol];
            Bv[kc].y = lds_x[(rl + 1) * D_ + bcol];
        }
        // 16-deep fp32 WMMA accumulation chain (K = 64 per s-chunk).
#pragma unroll
        for (int kc = 0; kc < 16; ++kc) {
            acc = __builtin_amdgcn_wmma_f32_16x16x4_f32(
                false, A[kc], false, Bv[kc], (short)0, acc, false, false);
        }
    }
    __syncthreads();

    // Apply softmax denominator (row scale) and park tile in LDS.
    const float* invZh = invZ + h * TP;
#pragma unroll
    for (int r = 0; r < 8; ++r) {
        const int m  = r + half * 8;                    // C/D layout row
        const int tr = t0 + tsub * 16 + m;
        const float z = (tr <= T_) ? invZh[tr] : 0.0f;
        lds_out[(tsub * 16 + m) * 68 + dsub * 16 + l15] = acc[r] * z;
    }
    __syncthreads();

    // Double l2norm per row: scale = 1/((n+eps) * (n/(n+eps)+eps)).
    if (tid < 32) {
        float ss = 0.0f;
#pragma unroll
        for (int j = 0; j < D_; ++j) {
            float v = lds_out[tid * 68 + j];
            ss += v * v;
        }
        const float n  = sqrtf(ss);
        const float s1 = n + EPSN;
        const float n2 = n / s1;
        const float s2 = n2 + EPSN;
        lds_scale[tid] = 1.0f / (s1 * s2);
    }
    __syncthreads();

    // Vectorized store of the 32x64 tile (guard past t=2048).
    for (int i = tid; i < 32 * (D_ / 4); i += 256) {
        const int r  = i >> 4;
        const int c4 = i & 15;
        const int tr = t0 + r;
        if (tr < TP) {
            float4 v = ((const float4*)lds_out)[r * 17 + c4];
            const float sc4 = lds_scale[r];
            v.x *= sc4; v.y *= sc4; v.z *= sc4; v.w *= sc4;
            ((float4*)out)[(((size_t)h * B_ + b) * TP + tr) * (D_ / 4) + c4] = v;
        }
    }
}

// ---------------------------------------------------------------------------
// Kernel 3: channel H = double-l2norm of the padded input itself.
// Block = 256 threads = 4 rows x 64 lanes; tree reduction in LDS.
// ---------------------------------------------------------------------------
__global__ __launch_bounds__(256)
void xp_norm(const float* __restrict__ x, float* __restrict__ out) {
    __shared__ float red[256];
    const int row = blockIdx.x * 4 + (threadIdx.x >> 6);   // in [0, B*TP)
    const int j   = threadIdx.x & 63;
    const int b   = row / TP;
    const int t   = row % TP;
    float v = 0.0f;
    if (t < T_) v = x[((size_t)b * T_ + t) * D_ + j];
    red[threadIdx.x] = v * v;
    __syncthreads();
    const int g = threadIdx.x & ~63;
    for (int off = 32; off > 0; off >>= 1) {
        if (j < off) red[threadIdx.x] += red[threadIdx.x + off];
        __syncthreads();
    }
    const float n  = sqrtf(red[g]);
    const float s1 = n + EPSN;
    const float n2 = n / s1;
    const float s2 = n2 + EPSN;
    out[(((size_t)H_ * B_ + b) * TP + t) * D_ + j] = v * (1.0f / (s1 * s2));
}

extern "C" void kernel_launch(void* const* d_in, const int* in_sizes, int n_in,
                              void* d_out, int out_size, void* d_ws, size_t ws_size,
                              hipStream_t stream) {
    const float* x = (const float*)d_in[0];   // [8, 2048, 64]
    const float* W = (const float*)d_in[1];   // [2048, 8]
    float* out  = (float*)d_out;              // [9, 8, 2049, 64]
    float* eW   = (float*)d_ws;               // [8][2049]
    float* invZ = eW + H_ * TP;               // [8][2049]  (~128 KB total scratch)

    toeplitz_precompute<<<H_, 256, 0, stream>>>(W, eW, invZ);

    dim3 grid((TP + 31) / 32, B_, H_);        // 65 x 8 x 8
    toeplitz_attn<<<grid, 256, 0, stream>>>(x, eW, invZ, out);

    xp_norm<<<(B_ * TP) / 4, 256, 0, stream>>>(x, out);
}